// GatedBlock_15753940042087
// MI455X (gfx1250) — compile-verified
//
#include <hip/hip_runtime.h>
#include <hip/hip_bf16.h>
#include <stdint.h>

typedef __attribute__((ext_vector_type(2))) float v2f;
typedef __attribute__((ext_vector_type(8))) float v8f;

#define XCOLS   960
#define OUTCOLS 768
#define ROWS_PER_WG 32   // two 16-row M-tiles; B-fragments reused 2x from regs

__device__ __forceinline__ v8f wmma_f32(v2f a, v2f b, v8f c) {
    // D = A(16x4 f32) * B(4x16 f32) + C(16x16 f32), full fp32 precision
    return __builtin_amdgcn_wmma_f32_16x16x4_f32(
        false, a, false, b, (short)0, c, false, false);
}

__device__ __forceinline__ unsigned lds_byte_addr(const void* p) {
    // generic pointer to LDS: low 32 bits are the LDS byte offset (ISA 10.2)
    return (unsigned)(uintptr_t)p;
}

__device__ __forceinline__ float fast_sigmoid(float v) {
    // v_exp_f32 + v_rcp_f32 (avoids the IEEE div_scale/div_fixup expansion)
    return __builtin_amdgcn_rcpf(1.0f + __expf(-v));
}

__global__ __launch_bounds__(256) void gated_block_kernel(
    const float* __restrict__ x,   // (n, 960)
    const float* __restrict__ W0,  // (256, 384)
    const float* __restrict__ W1,  // (128, 64)
    const float* __restrict__ W2,  // (64, 64)
    float* __restrict__ out,       // (n, 768)
    int n)
{
    __shared__ float lds_x[ROWS_PER_WG * XCOLS]; // 122880 B: staged x tile
    __shared__ float lds_g[ROWS_PER_WG * 128];   //  16384 B: sigmoid gates

    const int tid  = threadIdx.x;
    const int wave = tid >> 5;
    const int lane = tid & 31;
    const int half = lane >> 4;   // 0: K={0,1}, 1: K={2,3}
    const int l16  = lane & 15;
    const int row0 = blockIdx.x * ROWS_PER_WG;
    const int rowlim = n - row0;              // >= 32 for full tiles
    const bool full = (rowlim >= ROWS_PER_WG);

    // ---- Stage x tile -> LDS via CDNA5 async copy (ASYNCcnt) ----
    // 32 rows x 240 float4 = 7680 b128 transfers; 256 threads x 30 iters.
    #pragma unroll
    for (int i = 0; i < 30; ++i) {
        int idx  = i * 256 + tid;
        int row  = idx / 240;
        int c4   = idx % 240;
        int grow = row0 + row;
        if (grow > n - 1) grow = n - 1;                 // clamp tail tile
        const float* gp = x + (size_t)grow * XCOLS + c4 * 4;
        unsigned la = lds_byte_addr(&lds_x[row * XCOLS + c4 * 4]);
        asm volatile("global_load_async_to_lds_b128 %0, %1, off"
                     :: "v"(la), "v"(gp) : "memory");
    }
    asm volatile("s_wait_asynccnt 0" ::: "memory");
    __syncthreads();

    // ---- Phase 1: y0 = x0 @ W0, 24 N-tiles x 2 M-tiles, 3 N-tiles/wave ----
    {
        v8f acc[2][3] = {};
        const int kk0 = 2 * half; // this half-wave's K sub-offset
        #pragma unroll 2
        for (int k = 0; k < 256; k += 4) {
            // A fragments for both 16-row M-tiles (ds_load_b64 each)
            v2f a0 = *(const v2f*)&lds_x[(     l16) * XCOLS + k + kk0];
            v2f a1 = *(const v2f*)&lds_x[(16 + l16) * XCOLS + k + kk0];
            #pragma unroll
            for (int t = 0; t < 3; ++t) {
                int col = (3 * wave + t) * 16 + l16;
                v2f b;   // one B fragment feeds two WMMAs
                b.x = W0[(size_t)(k + kk0 + 0) * 384 + col];
                b.y = W0[(size_t)(k + kk0 + 1) * 384 + col];
                acc[0][t] = wmma_f32(a0, b, acc[0][t]);
                acc[1][t] = wmma_f32(a1, b, acc[1][t]);
            }
        }
        const float scale0 = 0.0625f; // 1/sqrt(256)
        #pragma unroll
        for (int m = 0; m < 2; ++m) {
            #pragma unroll
            for (int t = 0; t < 3; ++t) {
                int nt  = 3 * wave + t;
                int col = nt * 16 + l16;
                #pragma unroll
                for (int r = 0; r < 8; ++r) {
                    int row = m * 16 + r + 8 * half;
                    float v   = acc[m][t][r] * scale0;
                    float sig = fast_sigmoid(v);
                    if (nt < 16) { // scalars: silu * SILU_NORM
                        float o = v * sig * 1.6791f;
                        if (full) {
                            out[(size_t)(row0 + row) * OUTCOLS + col] = o;
                        } else if (row < rowlim) {
                            out[(size_t)(row0 + row) * OUTCOLS + col] = o;
                        }
                    } else {       // gates: sigmoid * SIGMOID_NORM -> LDS
                        lds_g[row * 128 + (col - 256)] = sig * 1.8484f;
                    }
                }
            }
        }
    }
    __syncthreads();

    // ---- Phase 2: y1 (12 tiles) + y2 (20 tiles) x 2 M-tiles, 4/wave ----
    #pragma unroll
    for (int tt = 0; tt < 4; ++tt) {
        const int tile = wave * 4 + tt;
        int ktile, ch, K, xbase, cstride, gofs;
        float scale;
        const float* __restrict__ W;
        if (tile < 12) {           // y1: 4 ktiles x 3 channels, K=128
            ktile = tile / 3; ch = tile % 3;
            K = 128; xbase = 256; cstride = 3; gofs = 0;
            W = W1; scale = 0.08838834764831845f;      // 1/sqrt(128)
        } else {                   // y2: 4 ktiles x 5 channels, K=64
            int t2 = tile - 12;
            ktile = t2 / 5; ch = t2 % 5;
            K = 64; xbase = 640; cstride = 5; gofs = 64;
            W = W2; scale = 0.125f;                    // 1/sqrt(64)
        }
        const int col = ktile * 16 + l16;
        const int kk0 = 2 * half;
        v8f acc0 = {}, acc1 = {};
        #pragma unroll 2
        for (int k = 0; k < K; k += 4) {
            int kk = k + kk0;
            int ao = xbase + (kk + 0) * cstride + ch;
            v2f a0, a1, b;
            a0.x = lds_x[(     l16) * XCOLS + ao];
            a0.y = lds_x[(     l16) * XCOLS + ao + cstride];
            a1.x = lds_x[(16 + l16) * XCOLS + ao];
            a1.y = lds_x[(16 + l16) * XCOLS + ao + cstride];
            b.x  = W[(size_t)(kk + 0) * 64 + col];
            b.y  = W[(size_t)(kk + 1) * 64 + col];
            acc0 = wmma_f32(a0, b, acc0);
            acc1 = wmma_f32(a1, b, acc1);
        }
        const int kout = ktile * 16 + l16;
        const int ocol = (tile < 12) ? (256 + kout * 3 + ch)
                                     : (448 + kout * 5 + ch);
        #pragma unroll
        for (int m = 0; m < 2; ++m) {
            #pragma unroll
            for (int r = 0; r < 8; ++r) {
                int row = m * 16 + r + 8 * half;
                float g = lds_g[row * 128 + gofs + kout];
                float o = (m ? acc1[r] : acc0[r]) * scale * g;
                if (full) {
                    out[(size_t)(row0 + row) * OUTCOLS + ocol] = o;
                } else if (row < rowlim) {
                    out[(size_t)(row0 + row) * OUTCOLS + ocol] = o;
                }
            }
        }
    }
}

extern "C" void kernel_launch(void* const* d_in, const int* in_sizes, int n_in,
                              void* d_out, int out_size, void* d_ws, size_t ws_size,
                              hipStream_t stream) {
    const float* x  = (const float*)d_in[0];
    const float* W0 = (const float*)d_in[1];
    const float* W1 = (const float*)d_in[2];
    const float* W2 = (const float*)d_in[3];
    float* out = (float*)d_out;
    const int n = in_sizes[0] / XCOLS;
    const int grid = (n + ROWS_PER_WG - 1) / ROWS_PER_WG; // 6250 for N=200000
    hipLaunchKernelGGL(gated_block_kernel, dim3(grid), dim3(256), 0, stream,
                       x, W0, W1, W2, out, n);
}